// DL_GNN_73564199846204
// MI455X (gfx1250) — compile-verified
//
#include <hip/hip_runtime.h>
#include <hip/hip_bf16.h>

// GCN forward on MI455X (gfx1250, wave32).
//
// Pipeline (all on `stream`, scratch carved from d_ws):
//  1) in-degree counts (int atomics)        -> counts[N]
//  2) dis[i] = rsqrt(counts[i]+1)           (deg includes self-loop)
//  3) exclusive scan of counts              -> rowptr[N+1]  (CSR by dst)
//  4) CSR fill (atomic cursors)             -> colidx[E]
//  5) Hs1 = (X @ W1) * dis[row]             [WMMA f32 16x16x4, 4 N-tiles/wave]
//  6) A1  = relu(dis*(CSR-sum Hs1 + Hs1[i]) + b1)   (wave-per-node gather, 4x MLP unroll)
//  7) Hs2 = (A1 @ W2) * dis[row]            [WMMA]
//  8) A2  = relu(dis*(CSR-sum Hs2 + Hs2[i]) + b2)
//  9) column means of A2 -> gsum[64]; out = gsum/N @ fcW + fcb

typedef __attribute__((ext_vector_type(2))) float v2f;
typedef __attribute__((ext_vector_type(8))) float v8f;

#define THREADS 256

// ---------------------------------------------------------------- degree
__global__ void k_count(const int* __restrict__ dst, int E, int* __restrict__ counts) {
    int e = blockIdx.x * blockDim.x + threadIdx.x;
    if (e < E) atomicAdd(&counts[dst[e]], 1);
}

// dis[i] = deg^{-1/2}, deg = in-edges + self-loop (always >= 1)
__global__ void k_dis(const int* __restrict__ counts, int n, float* __restrict__ dis) {
    int i = blockIdx.x * blockDim.x + threadIdx.x;
    if (i < n) dis[i] = rsqrtf((float)(counts[i] + 1));
}

// ---------------------------------------------------------------- scan (2048 elems / block)
__global__ void k_scan1(const int* __restrict__ counts, int n, int* __restrict__ blockSums) {
    __shared__ int sdata[THREADS];
    int base = blockIdx.x * 2048 + threadIdx.x * 8;
    int s = 0;
#pragma unroll
    for (int j = 0; j < 8; j++) { int idx = base + j; if (idx < n) s += counts[idx]; }
    sdata[threadIdx.x] = s;
    __syncthreads();
    for (int off = THREADS / 2; off > 0; off >>= 1) {
        if (threadIdx.x < off) sdata[threadIdx.x] += sdata[threadIdx.x + off];
        __syncthreads();
    }
    if (threadIdx.x == 0) blockSums[blockIdx.x] = sdata[0];
}

__global__ void k_scan2(int* __restrict__ blockSums, int B) {
    if (blockIdx.x == 0 && threadIdx.x == 0) {
        int acc = 0;
        for (int i = 0; i < B; i++) { int v = blockSums[i]; blockSums[i] = acc; acc += v; }
    }
}

__global__ void k_scan3(const int* __restrict__ counts, int n,
                        const int* __restrict__ blockOff, int* __restrict__ rowptr, int E) {
    __shared__ int tsum[THREADS];
    int tid = threadIdx.x;
    int base = blockIdx.x * 2048 + tid * 8;
    int v[8];
    int s = 0;
#pragma unroll
    for (int j = 0; j < 8; j++) {
        int idx = base + j;
        int t = (idx < n) ? counts[idx] : 0;
        v[j] = s;            // local exclusive prefix
        s += t;
    }
    tsum[tid] = s;
    __syncthreads();
    // Hillis-Steele inclusive scan of per-thread totals
    for (int off = 1; off < THREADS; off <<= 1) {
        int t = (tid >= off) ? tsum[tid - off] : 0;
        __syncthreads();
        tsum[tid] += t;
        __syncthreads();
    }
    int texcl = tsum[tid] - s;
    int boff = blockOff[blockIdx.x];
#pragma unroll
    for (int j = 0; j < 8; j++) {
        int idx = base + j;
        if (idx < n) rowptr[idx] = boff + texcl + v[j];
    }
    if (blockIdx.x == 0 && tid == 0) rowptr[n] = E;
}

// ---------------------------------------------------------------- CSR fill
__global__ void k_fill(const int* __restrict__ src, const int* __restrict__ dst, int E,
                       const int* __restrict__ rowptr, int* __restrict__ cursor,
                       int* __restrict__ colidx) {
    int e = blockIdx.x * blockDim.x + threadIdx.x;
    if (e < E) {
        int d = dst[e];
        int pos = atomicAdd(&cursor[d], 1);
        colidx[rowptr[d] + pos] = src[e];
    }
}

// ---------------------------------------------------------------- WMMA GEMM:  out = (A[MxK] @ W[Kx64]) * dis[row]
// One wave per 16-row stripe; each wave owns all four 16x16 N-tiles so the A
// fragment is loaded ONCE per k-step and reused 4x (no redundant A traffic).
// The 4 WMMAs per k-step hit different accumulators -> no RAW hazard, full pipelining.
// A-frag (f32 16x4, ISA 7.12.2): lanes 0-15 hold K=0,1 (v0,v1); lanes 16-31 hold K=2,3.
// B-frag (f32 4x16): mirrored — lanes 0-15 rows K=0,1; lanes 16-31 rows K=2,3; lane%16 = N.
// C/D (f32 16x16): VGPR j, lanes 0-15 -> M=j; lanes 16-31 -> M=j+8; lane%16 = N.
__global__ void __launch_bounds__(256)
k_gemm_wmma(const float* __restrict__ A, const float* __restrict__ W,
            const float* __restrict__ dis, float* __restrict__ out, int M, int K) {
    const int ln = threadIdx.x & 31;
    const int stripe = blockIdx.x * (blockDim.x >> 5) + (threadIdx.x >> 5);
    const int row0 = stripe * 16;
    if (row0 >= M) return;                        // wave-uniform

    if (row0 + 16 <= M) {                         // wave-uniform: EXEC all-ones on WMMA path
        const int mrow  = row0 + (ln & 15);
        const int khalf = (ln >> 4) * 2;          // 0 or 2
        const float* aptr = A + (size_t)mrow * K + khalf;
        const int bln = ln & 15;
        const float* bptr = W + (size_t)khalf * 64 + bln;

        v8f c0 = {0.f,0.f,0.f,0.f,0.f,0.f,0.f,0.f};
        v8f c1 = c0, c2 = c0, c3 = c0;

        for (int k = 0; k < K; k += 4) {
            v2f a;
            a.x = aptr[k];
            a.y = aptr[k + 1];
            const float* bk = bptr + (size_t)k * 64;
            v2f b0, b1, b2, b3;
            b0.x = bk[0];   b0.y = bk[64];
            b1.x = bk[16];  b1.y = bk[80];
            b2.x = bk[32];  b2.y = bk[96];
            b3.x = bk[48];  b3.y = bk[112];
            c0 = __builtin_amdgcn_wmma_f32_16x16x4_f32(false, a, false, b0, (short)0, c0, false, false);
            c1 = __builtin_amdgcn_wmma_f32_16x16x4_f32(false, a, false, b1, (short)0, c1, false, false);
            c2 = __builtin_amdgcn_wmma_f32_16x16x4_f32(false, a, false, b2, (short)0, c2, false, false);
            c3 = __builtin_amdgcn_wmma_f32_16x16x4_f32(false, a, false, b3, (short)0, c3, false, false);
        }

        const int rbase = row0 + (ln >> 4) * 8;
        float dsc[8];
#pragma unroll
        for (int j = 0; j < 8; j++) dsc[j] = dis[rbase + j];
#pragma unroll
        for (int j = 0; j < 8; j++) {
            float* orow = out + (size_t)(rbase + j) * 64 + bln;
            orow[0]  = c0[j] * dsc[j];
            orow[16] = c1[j] * dsc[j];
            orow[32] = c2[j] * dsc[j];
            orow[48] = c3[j] * dsc[j];
        }
    } else {                                      // ragged tail stripe: VALU fallback
        int rem = M - row0;                       // 1..15
        for (int t = ln; t < rem * 64; t += 32) {
            int r = t >> 6, col = t & 63;
            const float* ar = A + (size_t)(row0 + r) * K;
            float s = 0.f;
            for (int k = 0; k < K; k++) s = fmaf(ar[k], W[(size_t)k * 64 + col], s);
            out[(size_t)(row0 + r) * 64 + col] = s * dis[row0 + r];
        }
    }
}

// ---------------------------------------------------------------- aggregation: wave per node
// out[i] = relu( dis[i] * ( sum_{e: dst=i} Hs[src_e] + Hs[i] ) + bias )
// 4x unroll with independent accumulators: 4 row-gathers in flight per wave.
__global__ void k_aggregate(const float* __restrict__ Hs, const int* __restrict__ rowptr,
                            const int* __restrict__ colidx, const float* __restrict__ dis,
                            const float* __restrict__ bias, float* __restrict__ out, int n) {
    int wid = (blockIdx.x * blockDim.x + threadIdx.x) >> 5;   // wave32 -> node id
    int ln = threadIdx.x & 31;                                // lane -> channels 2*ln, 2*ln+1
    if (wid >= n) return;

    int beg = rowptr[wid], end = rowptr[wid + 1];
    float2 sv = ((const float2*)(Hs + (size_t)wid * 64))[ln];
    float ax = sv.x, ay = sv.y;
    float bx = 0.f, by = 0.f, cx = 0.f, cy = 0.f, ex = 0.f, ey = 0.f;

    int j = beg;
    for (; j + 4 <= end; j += 4) {
        if ((((j - beg) & 63) == 0) && (j + 128 < end))
            __builtin_prefetch(&colidx[j + 128], 0, 0);       // global_prefetch_b8
        int s0 = colidx[j];
        int s1 = colidx[j + 1];
        int s2 = colidx[j + 2];
        int s3 = colidx[j + 3];
        float2 v0 = ((const float2*)(Hs + (size_t)s0 * 64))[ln];
        float2 v1 = ((const float2*)(Hs + (size_t)s1 * 64))[ln];
        float2 v2 = ((const float2*)(Hs + (size_t)s2 * 64))[ln];
        float2 v3 = ((const float2*)(Hs + (size_t)s3 * 64))[ln];
        ax += v0.x; ay += v0.y;
        bx += v1.x; by += v1.y;
        cx += v2.x; cy += v2.y;
        ex += v3.x; ey += v3.y;
    }
    for (; j < end; j++) {
        int s = colidx[j];
        float2 v = ((const float2*)(Hs + (size_t)s * 64))[ln];
        ax += v.x; ay += v.y;
    }
    ax += bx + cx + ex;
    ay += by + cy + ey;

    float d = dis[wid];
    float2 bb = ((const float2*)bias)[ln];
    float2 o;
    o.x = fmaxf(fmaf(d, ax, bb.x), 0.f);
    o.y = fmaxf(fmaf(d, ay, bb.y), 0.f);
    ((float2*)(out + (size_t)wid * 64))[ln] = o;
}

// ---------------------------------------------------------------- readout
__global__ void k_colsum(const float* __restrict__ A, int n, float* __restrict__ gsum) {
    __shared__ float sh[THREADS * 2];
    int ln = threadIdx.x & 31, wv = threadIdx.x >> 5;
    float ax = 0.f, ay = 0.f;
    for (int r = blockIdx.x * 8 + wv; r < n; r += gridDim.x * 8) {
        float2 v = ((const float2*)(A + (size_t)r * 64))[ln];
        ax += v.x;
        ay += v.y;
    }
    sh[threadIdx.x * 2] = ax;
    sh[threadIdx.x * 2 + 1] = ay;
    __syncthreads();
    if (wv == 0) {
        for (int w = 1; w < 8; w++) {
            ax += sh[(w * 32 + ln) * 2];
            ay += sh[(w * 32 + ln) * 2 + 1];
        }
        atomicAdd(&gsum[2 * ln], ax);
        atomicAdd(&gsum[2 * ln + 1], ay);
    }
}

__global__ void k_fc(const float* __restrict__ gsum, const float* __restrict__ fcW,
                     const float* __restrict__ fcb, float* __restrict__ out, float invN) {
    int t = threadIdx.x;
    if (t < 2) {
        float s = fcb[t];
        for (int c = 0; c < 64; c++) s = fmaf(gsum[c] * invN, fcW[c * 2 + t], s);
        out[t] = s;
    }
}

// ---------------------------------------------------------------- launch
extern "C" void kernel_launch(void* const* d_in, const int* in_sizes, int n_in,
                              void* d_out, int out_size, void* d_ws, size_t ws_size,
                              hipStream_t stream) {
    const float* x   = (const float*)d_in[0];
    const int*   ei  = (const int*)d_in[1];
    const float* W1  = (const float*)d_in[2];
    const float* b1  = (const float*)d_in[3];
    const float* W2  = (const float*)d_in[4];
    const float* b2  = (const float*)d_in[5];
    const float* fcW = (const float*)d_in[6];
    const float* fcb = (const float*)d_in[7];

    const int N = in_sizes[0] / 128;    // 100000
    const int E = in_sizes[1] / 2;      // 3200000
    const int* src = ei;                // edge_index[0]
    const int* dst = ei + E;            // edge_index[1]

    // carve workspace (256B aligned)
    char* w = (char*)d_ws;
    auto carve = [&](size_t bytes) -> void* {
        void* p = (void*)w;
        w += (bytes + 255) & ~(size_t)255;
        return p;
    };
    int*   counts    = (int*)carve((size_t)N * 4);          // reused as cursor after scan
    int*   rowptr    = (int*)carve((size_t)(N + 1) * 4);
    float* dis       = (float*)carve((size_t)N * 4);
    float* gsum      = (float*)carve(64 * 4);
    int*   blockSums = (int*)carve(4096);
    int*   colidx    = (int*)carve((size_t)E * 4);
    float* bufA      = (float*)carve((size_t)N * 64 * 4);   // Hs1, then Hs2
    float* bufB      = (float*)carve((size_t)N * 64 * 4);   // A1,  then A2

    hipMemsetAsync(counts, 0, (size_t)N * 4, stream);
    hipMemsetAsync(gsum, 0, 64 * 4, stream);

    // graph structure
    k_count<<<(E + THREADS - 1) / THREADS, THREADS, 0, stream>>>(dst, E, counts);
    k_dis<<<(N + THREADS - 1) / THREADS, THREADS, 0, stream>>>(counts, N, dis);
    int SB = (N + 2047) / 2048;
    k_scan1<<<SB, THREADS, 0, stream>>>(counts, N, blockSums);
    k_scan2<<<1, 32, 0, stream>>>(blockSums, SB);
    k_scan3<<<SB, THREADS, 0, stream>>>(counts, N, blockSums, rowptr, E);
    hipMemsetAsync(counts, 0, (size_t)N * 4, stream);       // counts -> cursor
    k_fill<<<(E + THREADS - 1) / THREADS, THREADS, 0, stream>>>(src, dst, E, rowptr, counts, colidx);

    const int STRIPES = (N + 15) / 16;              // 16-row stripes
    const int GBLK = (STRIPES + 7) / 8;             // 8 waves (stripes) per block
    // layer 1
    k_gemm_wmma<<<GBLK, 256, 0, stream>>>(x, W1, dis, bufA, N, 128);
    k_aggregate<<<(N + 7) / 8, THREADS, 0, stream>>>(bufA, rowptr, colidx, dis, b1, bufB, N);
    // layer 2
    k_gemm_wmma<<<GBLK, 256, 0, stream>>>(bufB, W2, dis, bufA, N, 64);
    k_aggregate<<<(N + 7) / 8, THREADS, 0, stream>>>(bufA, rowptr, colidx, dis, b2, bufB, N);
    // readout
    k_colsum<<<240, THREADS, 0, stream>>>(bufB, N, gsum);
    k_fc<<<1, 32, 0, stream>>>(gsum, fcW, fcb, (float*)d_out, 1.0f / (float)N);
}